// AffinityLoss_34986803593896
// MI455X (gfx1250) — compile-verified
//
#include <hip/hip_runtime.h>
#include <hip/hip_bf16.h>

// Problem constants (match reference)
#define NB    4
#define NUMC  21
#define HH    768
#define WW    768
#define NH    96
#define NW    96
#define HWF   (HH*WW)      // 589824
#define HW2   (NH*NW)      // 9216
#define BOT_EPS 1e-4f
#define KLD_MARGIN 3.0f

typedef __attribute__((ext_vector_type(2))) float v2f;
typedef __attribute__((ext_vector_type(8))) float v8f;

// ---------------------------------------------------------------------------
// Exact f32 wave32 allreduce using V_WMMA_F32_16X16X4_F32 (two chained WMMAs).
// B = all-ones so B's VGPR layout is irrelevant. A layout (documented):
//   lane m (0..15):  row m, K=0 in VGPR0 ; lane m+16: row m, K=2 in VGPR0.
// WMMA1: D[m][n] = s[m] + s[m+16]. D layout: VGPR r = rowsum(r) for lanes
// 0..15, rowsum(r+8) for lanes 16..31  =>  per-lane sum of the 8 D regs is
// S_lo (lanes<16) or S_hi (lanes>=16). WMMA2 on that value gives
// D2[m][n] = S_lo + S_hi = TOTAL in every element. EXEC must be all ones.
// ---------------------------------------------------------------------------
__device__ __forceinline__ float wave_allreduce_sum(float x) {
  v2f a; a[0] = x; a[1] = 0.0f;
  v2f ones; ones[0] = 1.0f; ones[1] = 1.0f;
  v8f c = {};
  v8f d = __builtin_amdgcn_wmma_f32_16x16x4_f32(false, a, false, ones,
                                                (short)0, c, false, false);
  float t = ((d[0] + d[1]) + (d[2] + d[3])) + ((d[4] + d[5]) + (d[6] + d[7]));
  v2f a2; a2[0] = t; a2[1] = 0.0f;
  v8f d2 = __builtin_amdgcn_wmma_f32_16x16x4_f32(false, a2, false, ones,
                                                 (short)0, c, false, false);
  return d2[0];
}

// ---------------------------------------------------------------------------
// Kernel 0: zero the accumulators (graph-replay safe re-init every launch)
// acc[0]=nll_sum acc[1]=valid_cnt acc[2]=edge_num acc[3]=edge_cnt
// acc[4]=ne_num  acc[5]=ne_cnt
// ---------------------------------------------------------------------------
__global__ void zero_acc_kernel(float* acc) {
  if (threadIdx.x < 8) acc[threadIdx.x] = 0.0f;
}

// ---------------------------------------------------------------------------
// Kernel 1: full-res cross entropy (memory-bound: one pass over 198 MB).
// One thread per 4-pixel group: 21 x global_load_b128 (512B/wave/channel).
// WMMA wave allreduce -> LDS block reduce -> 1 atomic per block per acc.
// ---------------------------------------------------------------------------
__global__ __launch_bounds__(256)
void ce_kernel(const float* __restrict__ logits, const int* __restrict__ labels,
               float* __restrict__ acc) {
  const int HW4 = HWF / 4;            // 147456 groups per image
  const int total4 = NB * HW4;        // 589824 groups
  float nll_sum = 0.0f, cnt = 0.0f;

  for (int g = blockIdx.x * blockDim.x + threadIdx.x; g < total4;
       g += gridDim.x * blockDim.x) {
    int n = g / HW4;
    int hw4 = g - n * HW4;
    int hw = hw4 * 4;
    const float4* base =
        (const float4*)(logits + (size_t)n * NUMC * HWF + hw);
    float rr[4][NUMC];
#pragma unroll
    for (int cc = 0; cc < NUMC; ++cc) {
      float4 t = base[(size_t)cc * (HWF / 4)];
      rr[0][cc] = t.x; rr[1][cc] = t.y; rr[2][cc] = t.z; rr[3][cc] = t.w;
    }
    int4 lab4 = *(const int4*)(labels + (size_t)n * HWF + hw);
    int labs[4] = {lab4.x, lab4.y, lab4.z, lab4.w};
#pragma unroll
    for (int s = 0; s < 4; ++s) {
      int lab = labs[s];
      bool valid = (lab != 255);
      int sl = valid ? lab : 0;
      sl = min(max(sl, 0), NUMC - 1);
      float m = rr[s][0];
#pragma unroll
      for (int cc = 1; cc < NUMC; ++cc) m = fmaxf(m, rr[s][cc]);
      float ssum = 0.0f;
#pragma unroll
      for (int cc = 0; cc < NUMC; ++cc) ssum += __expf(rr[s][cc] - m);
      float nll = (m + __logf(ssum)) - rr[s][sl];
      if (valid) { nll_sum += nll; cnt += 1.0f; }
    }
  }

  // wave (WMMA) -> block (LDS) -> global (single atomic per block)
  __shared__ float lds0[8], lds1[8];
  float w0 = wave_allreduce_sum(nll_sum);
  float w1 = wave_allreduce_sum(cnt);
  int wid = threadIdx.x >> 5;
  if ((threadIdx.x & 31) == 0) { lds0[wid] = w0; lds1[wid] = w1; }
  __syncthreads();
  if (threadIdx.x == 0) {
    float a = 0.0f, b = 0.0f;
#pragma unroll
    for (int k = 0; k < 8; ++k) { a += lds0[k]; b += lds1[k]; }
    atomicAdd(&acc[0], a);
    atomicAdd(&acc[1], b);
  }
}

// ---------------------------------------------------------------------------
// Kernel 2: downsample. One thread per (n,j,i) over 4x96x96 = 36864.
// Writes clipped masked softmax probs (N,C,96,96) and lab_ds (N,96,96).
// ---------------------------------------------------------------------------
__global__ __launch_bounds__(256)
void ds_kernel(const float* __restrict__ logits, const int* __restrict__ labels,
               float* __restrict__ probs, int* __restrict__ lab_ds) {
  int idx = blockIdx.x * blockDim.x + threadIdx.x;
  if (idx >= NB * HW2) return;
  int n = idx / HW2;
  int rem = idx - n * HW2;
  int j = rem / NW;
  int i = rem - j * NW;

  // nearest labels: iy = j*768//96 = j*8
  int lab = labels[n * HWF + (j * 8) * WW + (i * 8)];
  lab_ds[idx] = lab;

  // bilinear align_corners sample positions
  const float step = 767.0f / 95.0f;
  float fy = (float)j * step;
  float fx = (float)i * step;
  int y0 = (int)floorf(fy); y0 = min(max(y0, 0), HH - 1);
  int x0 = (int)floorf(fx); x0 = min(max(x0, 0), WW - 1);
  int y1 = min(y0 + 1, HH - 1);
  int x1 = min(x0 + 1, WW - 1);
  float wy = fy - (float)y0;
  float wx = fx - (float)x0;
  float w00 = (1.0f - wy) * (1.0f - wx);
  float w01 = (1.0f - wy) * wx;
  float w10 = wy * (1.0f - wx);
  float w11 = wy * wx;

  const float* b = logits + (size_t)n * NUMC * HWF;
  float v[NUMC];
#pragma unroll
  for (int c = 0; c < NUMC; ++c) {
    const float* pc = b + (size_t)c * HWF;
    v[c] = pc[y0 * WW + x0] * w00 + pc[y0 * WW + x1] * w01 +
           pc[y1 * WW + x0] * w10 + pc[y1 * WW + x1] * w11;
  }

  float maskf = (lab < NUMC) ? 1.0f : 0.0f;
  float m = v[0];
#pragma unroll
  for (int c = 1; c < NUMC; ++c) m = fmaxf(m, v[c]);
  float s = 0.0f;
#pragma unroll
  for (int c = 0; c < NUMC; ++c) s += __expf(v[c] - m);
  float inv = 1.0f / s;
#pragma unroll
  for (int c = 0; c < NUMC; ++c) {
    float p = __expf(v[c] - m) * inv * maskf;      // softmax * mask
    p = fminf(fmaxf(p, BOT_EPS), 1.0f);            // clip
    probs[(((size_t)n * NUMC + c) * NH + j) * NW + i] = p;
  }
}

// ---------------------------------------------------------------------------
// Kernel 3: affinity KL over 8 neighbors. One thread per (n,c,j,i) = 774144.
// Padding semantics: label pad=255 -> edge=1 ; prob pad=0 -> clip -> 1e-4.
// ---------------------------------------------------------------------------
__global__ __launch_bounds__(256)
void aff_kernel(const float* __restrict__ probs, const int* __restrict__ lab_ds,
                float* __restrict__ acc) {
  const int CHW = NUMC * HW2;
  const int total = NB * CHW;
  const int dy[8] = {-1, -1, -1, 0, 0, 1, 1, 1};
  const int dx[8] = {-1, 0, 1, -1, 1, -1, 0, 1};
  float e_num = 0.0f, e_cnt = 0.0f, ne_num = 0.0f, ne_cnt = 0.0f;

  for (int p = blockIdx.x * blockDim.x + threadIdx.x; p < total;
       p += gridDim.x * blockDim.x) {
    int n = p / CHW;
    int rem = p - n * CHW;
    int c = rem / HW2;
    int r2 = rem - c * HW2;
    int j = r2 / NW;
    int i = r2 - j * NW;

    int labc = lab_ds[n * HW2 + r2];
    float ohc = ((labc < NUMC) && (labc == c)) ? 1.0f : 0.0f;
    float pc = probs[p];
    float log_pc = __logf(pc);
    float ncv = 1.0f - pc + BOT_EPS;
    float log_nc = __logf(ncv);

#pragma unroll
    for (int k = 0; k < 8; ++k) {
      int jn = j + dy[k];
      int in = i + dx[k];
      bool inb = (jn >= 0) & (jn < NH) & (in >= 0) & (in < NW);
      float ohn, pn;
      if (inb) {
        int labn = lab_ds[n * HW2 + jn * NW + in];
        ohn = ((labn < NUMC) && (labn == c)) ? 1.0f : 0.0f;
        pn = probs[(((size_t)n * NUMC + c) * NH + jn) * NW + in];
      } else {
        ohn = 255.0f;      // pad label -> always an edge
        pn = BOT_EPS;      // pad prob 0 -> clipped
      }
      float e = (ohn != ohc) ? 1.0f : 0.0f;
      float np_ = 1.0f - pn + BOT_EPS;
      float kl = pn * (__logf(pn) - log_pc) + np_ * (__logf(np_) - log_nc);
      e_num  += fmaxf(0.0f, KLD_MARGIN - kl) * e;
      e_cnt  += e;
      ne_num += kl * (1.0f - e);
      ne_cnt += (1.0f - e);
    }
  }

  // wave (WMMA) -> block (LDS) -> global (single atomic per block per acc)
  __shared__ float lds[4][8];
  float s0 = wave_allreduce_sum(e_num);
  float s1 = wave_allreduce_sum(e_cnt);
  float s2 = wave_allreduce_sum(ne_num);
  float s3 = wave_allreduce_sum(ne_cnt);
  int wid = threadIdx.x >> 5;
  if ((threadIdx.x & 31) == 0) {
    lds[0][wid] = s0; lds[1][wid] = s1; lds[2][wid] = s2; lds[3][wid] = s3;
  }
  __syncthreads();
  if (threadIdx.x == 0) {
    float a0 = 0.0f, a1 = 0.0f, a2 = 0.0f, a3 = 0.0f;
#pragma unroll
    for (int k = 0; k < 8; ++k) {
      a0 += lds[0][k]; a1 += lds[1][k]; a2 += lds[2][k]; a3 += lds[3][k];
    }
    atomicAdd(&acc[2], a0);
    atomicAdd(&acc[3], a1);
    atomicAdd(&acc[4], a2);
    atomicAdd(&acc[5], a3);
  }
}

// ---------------------------------------------------------------------------
// Kernel 4: final scalar combine
// ---------------------------------------------------------------------------
__global__ void final_kernel(const float* __restrict__ acc,
                             const int* __restrict__ step,
                             float* __restrict__ out) {
  double nll = (double)acc[0] / fmax((double)acc[1], 1.0);
  double el  = (double)acc[2] / fmax((double)acc[3], 1.0);
  double nel = (double)acc[4] / fmax((double)acc[5], 1.0);
  double decay = exp(-((double)step[0]) / 30000.0 * log(20.0));
  out[0] = (float)(nll + decay * (el * 1.0 + nel * 1.0));
}

// ---------------------------------------------------------------------------
extern "C" void kernel_launch(void* const* d_in, const int* in_sizes, int n_in,
                              void* d_out, int out_size, void* d_ws, size_t ws_size,
                              hipStream_t stream) {
  const float* logits = (const float*)d_in[0];
  const int* labels   = (const int*)d_in[1];
  const int* step     = (const int*)d_in[2];
  float* out          = (float*)d_out;

  // workspace layout
  float* acc    = (float*)d_ws;                                   // 8 floats
  float* probs  = (float*)((char*)d_ws + 256);                    // 4*21*96*96 f32
  int*   lab_ds = (int*)((char*)d_ws + 256 +
                         (size_t)NB * NUMC * HW2 * sizeof(float)); // 4*96*96 i32

  zero_acc_kernel<<<1, 32, 0, stream>>>(acc);

  // CE: 589824 pixel-groups of 4 = 1152 blocks x 256 threads x 2 groups
  ce_kernel<<<1152, 256, 0, stream>>>(logits, labels, acc);

  // downsample: 36864 threads = 144 blocks x 256
  ds_kernel<<<144, 256, 0, stream>>>(logits, labels, probs, lab_ds);

  // affinity: 774144 = 3024 blocks x 256 exactly
  aff_kernel<<<3024, 256, 0, stream>>>(probs, lab_ds, acc);

  final_kernel<<<1, 1, 0, stream>>>(acc, step, out);
}